// StructuralCausalModel_37950331027880
// MI455X (gfx1250) — compile-verified
//
#include <hip/hip_runtime.h>
#include <math.h>

// ---------------------------------------------------------------------------
// StructuralCausalModel forward for MI455X (gfx1250, wave32, WMMA).
//
// Sequential scan over 3 layers x 32 vars -> 96 step kernels + 1 init kernel.
// d_out doubles as the live `values` (B=512, n=32, d=256) fp32 buffer.
// All GEMMs use V_WMMA_F32_16X16X4_F32 (full fp32, matches reference dtype).
// ---------------------------------------------------------------------------

typedef __attribute__((ext_vector_type(2))) float v2f;
typedef __attribute__((ext_vector_type(8))) float v8f;

#define N_VARS  32
#define D_MODEL 256
#define D_MECH  128
#define D_NOISE 64
#define BATCH   512
#define ROWS    16     // batch rows per block (one 16-row WMMA M-tile)
#define LDW     260    // padded LDS stride for 256-wide activations (bank-safe)
#define LDN     68     // padded LDS stride for 64-wide noise tile

__device__ __forceinline__ float gelu_exact(float x) {
    // exact-erf GELU, matches torch nn.GELU default / jax approximate=False
    return 0.5f * x * (1.0f + erff(x * 0.70710678118654752440f));
}

// One 16x16 fp32 output tile: A (16xK) from LDS, B (KxN slab) from global/L2.
// A-fragment layout (ISA 7.12.2, 32-bit A 16x4): lane l holds
//   m = l%16, k = kb + 2*(l/16) + {0,1}  -> contiguous float2 in LDS.
// B-fragment mirrors with n = l%16 (caller pre-offsets Bg by column n).
__device__ __forceinline__ v8f wmma_k_tile(const float* As, int lda,
                                           const float* Bg, int ldb,
                                           int K, v8f acc) {
    const int lane = threadIdx.x & 31;
    const int ml   = lane & 15;
    const int kl   = (lane >> 4) << 1;   // 0 or 2
#pragma unroll 4
    for (int kb = 0; kb < K; kb += 4) {
        const float* ap = As + ml * lda + kb + kl;
        v2f a; a.x = ap[0]; a.y = ap[1];
        const float* bp = Bg + (kb + kl) * ldb;
        v2f b; b.x = bp[0]; b.y = bp[ldb];
        acc = __builtin_amdgcn_wmma_f32_16x16x4_f32(
            /*neg_a=*/false, a, /*neg_b=*/false, b,
            /*c_mod=*/(short)0, acc, /*reuse_a=*/false, /*reuse_b=*/false);
    }
    return acc;
}

__global__ void scm_zero_kernel(float* __restrict__ p, int n4) {
    float4 z = make_float4(0.f, 0.f, 0.f, 0.f);
    for (int i = blockIdx.x * blockDim.x + threadIdx.x; i < n4;
         i += gridDim.x * blockDim.x)
        ((float4*)p)[i] = z;
}

__global__ __launch_bounds__(256)
void scm_var_step(const float* __restrict__ noise,        // (B, n, dn)
                  const float* __restrict__ edge_logits,  // (n, n)
                  const float* __restrict__ var_emb,      // (n, d)
                  const float* __restrict__ pa_W,         // (d, d)
                  const float* __restrict__ pa_b,         // (d)
                  const float* __restrict__ m_W1,         // (n, d, m)
                  const float* __restrict__ m_b1,         // (n, m)
                  const float* __restrict__ m_W2,         // (n, m, m)
                  const float* __restrict__ m_b2,         // (n, m)
                  const float* __restrict__ m_W3,         // (n, m, d)
                  const float* __restrict__ m_b3,         // (n, d)
                  const float* __restrict__ ne_W,         // (n, dn, d)
                  const float* __restrict__ ne_b,         // (n, d)
                  float* __restrict__ values,             // d_out, (B, n, d)
                  int var) {
    __shared__ float Xa[ROWS][LDW];
    __shared__ float Xb[ROWS][LDW];
    __shared__ float Nz[ROWS][LDN];
    __shared__ float sAdj[N_VARS];

    const int tid  = threadIdx.x;
    const int lane = tid & 31;
    const int wave = tid >> 5;
    const int b0   = blockIdx.x * ROWS;
    const int nlo  = lane & 15;          // column within a 16-wide tile
    const int mh   = (lane >> 4) * 8;    // C/D row base: m = r + 8*(lane/16)

    // Adjacency column i of adj = sigmoid(masked_logits / 0.5).
    // Diagonal is masked to logit 0 -> sigmoid(0) = 0.5 (NOT zero).
    if (tid < N_VARS) {
        float lg = (tid == var) ? 0.0f : edge_logits[tid * N_VARS + var];
        sAdj[tid] = 1.0f / (1.0f + expf(-2.0f * lg));
    }
    // Stage noise tile: Nz[r][:] = exogenous_noise[b0+r, var, :]  (16x64)
    {
        int r = tid >> 4, c4 = tid & 15;
        const float4* src =
            (const float4*)(noise + ((size_t)(b0 + r) * N_VARS + var) * D_NOISE);
        *((float4*)&Nz[r][0] + c4) = src[c4];
    }
    __syncthreads();

    // P0: Xa[b][:] = emb_i + sum_v adj[v][i] * values[b, v, :]
    for (int q = 0; q < 16; ++q) {
        int idx = tid + 256 * q;         // 64 rows x 64 float4 columns
        int r = idx >> 6, c4 = idx & 63;
        float4 acc = *((const float4*)(var_emb + (size_t)var * D_MODEL) + c4);
        const float* vb = values + (size_t)(b0 + r) * N_VARS * D_MODEL;
        for (int v = 0; v < N_VARS; ++v) {
            float4 x = *((const float4*)(vb + v * D_MODEL) + c4);
            float a = sAdj[v];
            acc.x += a * x.x; acc.y += a * x.y;
            acc.z += a * x.z; acc.w += a * x.w;
        }
        *((float4*)&Xa[r][0] + c4) = acc;
    }
    __syncthreads();

    // P1: Xb = gelu(Xa @ pa_W + pa_b)          (K=256, N=256)
    for (int ct = wave; ct < 16; ct += 8) {
        int nl = ct * 16 + nlo;
        v8f acc = {0.f,0.f,0.f,0.f,0.f,0.f,0.f,0.f};
        acc = wmma_k_tile(&Xa[0][0], LDW, pa_W + nl, D_MODEL, D_MODEL, acc);
        float bias = pa_b[nl];
#pragma unroll
        for (int r = 0; r < 8; ++r)
            Xb[mh + r][nl] = gelu_exact(acc[r] + bias);
    }
    __syncthreads();

    // P2: Xa[:,0:128] = gelu(Xb @ W1_i + b1_i) (K=256, N=128)
    {
        const float* W1 = m_W1 + (size_t)var * D_MODEL * D_MECH;
        int nl = wave * 16 + nlo;        // 8 waves cover N=128 exactly
        v8f acc = {0.f,0.f,0.f,0.f,0.f,0.f,0.f,0.f};
        acc = wmma_k_tile(&Xb[0][0], LDW, W1 + nl, D_MECH, D_MODEL, acc);
        float bias = m_b1[(size_t)var * D_MECH + nl];
#pragma unroll
        for (int r = 0; r < 8; ++r)
            Xa[mh + r][nl] = gelu_exact(acc[r] + bias);
    }
    __syncthreads();

    // P3: Xb[:,0:128] = gelu(Xa @ W2_i + b2_i) (K=128, N=128)
    {
        const float* W2 = m_W2 + (size_t)var * D_MECH * D_MECH;
        int nl = wave * 16 + nlo;
        v8f acc = {0.f,0.f,0.f,0.f,0.f,0.f,0.f,0.f};
        acc = wmma_k_tile(&Xa[0][0], LDW, W2 + nl, D_MECH, D_MECH, acc);
        float bias = m_b2[(size_t)var * D_MECH + nl];
#pragma unroll
        for (int r = 0; r < 8; ++r)
            Xb[mh + r][nl] = gelu_exact(acc[r] + bias);
    }
    __syncthreads();

    // P4: Xa = gelu(Nz @ ne_W_i + ne_b_i)      (K=64, N=256) noise encoder
    for (int ct = wave; ct < 16; ct += 8) {
        const float* NW = ne_W + (size_t)var * D_NOISE * D_MODEL;
        int nl = ct * 16 + nlo;
        v8f acc = {0.f,0.f,0.f,0.f,0.f,0.f,0.f,0.f};
        acc = wmma_k_tile(&Nz[0][0], LDN, NW + nl, D_MODEL, D_NOISE, acc);
        float bias = ne_b[(size_t)var * D_MODEL + nl];
#pragma unroll
        for (int r = 0; r < 8; ++r)
            Xa[mh + r][nl] = gelu_exact(acc[r] + bias);
    }
    __syncthreads();

    // P5: values[:, var, :] = Xb @ W3_i + b3_i + Xa   (K=128, N=256)
    for (int ct = wave; ct < 16; ct += 8) {
        const float* W3 = m_W3 + (size_t)var * D_MECH * D_MODEL;
        int nl = ct * 16 + nlo;
        v8f acc = {0.f,0.f,0.f,0.f,0.f,0.f,0.f,0.f};
        acc = wmma_k_tile(&Xb[0][0], LDW, W3 + nl, D_MODEL, D_MECH, acc);
        float bias = m_b3[(size_t)var * D_MODEL + nl];
#pragma unroll
        for (int r = 0; r < 8; ++r) {
            int m = mh + r;
            values[((size_t)(b0 + m) * N_VARS + var) * D_MODEL + nl] =
                acc[r] + bias + Xa[m][nl];
        }
    }
}

extern "C" void kernel_launch(void* const* d_in, const int* in_sizes, int n_in,
                              void* d_out, int out_size, void* d_ws, size_t ws_size,
                              hipStream_t stream) {
    const float* noise       = (const float*)d_in[0];
    const float* edge_logits = (const float*)d_in[1];
    const float* var_emb     = (const float*)d_in[2];
    const float* pa_W        = (const float*)d_in[3];
    const float* pa_b        = (const float*)d_in[4];
    const float* m_W1        = (const float*)d_in[5];
    const float* m_b1        = (const float*)d_in[6];
    const float* m_W2        = (const float*)d_in[7];
    const float* m_b2        = (const float*)d_in[8];
    const float* m_W3        = (const float*)d_in[9];
    const float* m_b3        = (const float*)d_in[10];
    const float* ne_W        = (const float*)d_in[11];
    const float* ne_b        = (const float*)d_in[12];
    float* out = (float*)d_out;
    (void)d_ws; (void)ws_size; (void)in_sizes; (void)n_in;

    // values must start at zero (reference scan begins from zeros)
    scm_zero_kernel<<<256, 256, 0, stream>>>(out, out_size / 4);

    // sequential scan: 3 layers x 32 vars, stream order provides the sync
    for (int layer = 0; layer < 3; ++layer) {
        for (int v = 0; v < N_VARS; ++v) {
            scm_var_step<<<BATCH / ROWS, 256, 0, stream>>>(
                noise, edge_logits, var_emb, pa_W, pa_b,
                m_W1, m_b1, m_W2, m_b2, m_W3, m_b3, ne_W, ne_b,
                out, v);
        }
    }
}